// HierarchicalBCEFocalLoss_32856499815101
// MI455X (gfx1250) — compile-verified
//
#include <hip/hip_runtime.h>
#include <hip/hip_bf16.h>

// HierarchicalBCEFocalLoss for MI455X (gfx1250, wave32).
// Memory-bound streaming reduction: 200MB in -> 1 float out.
// Wave reduction uses V_WMMA_F32_16X16X4_F32 (A = lane accumulators, B = ones).

typedef float v2f __attribute__((ext_vector_type(2)));
typedef float v8f __attribute__((ext_vector_type(8)));

#define L_DIM  25
#define BLOCK  256
#define NWAVES (BLOCK / 32)

// constexpr (NOT __constant__) so the fully-unrolled column loop folds these
// to straight-line code: parent saves become register copies, child checks
// become a single cmp+cndmask against a known register.
// Columns 0,4,7,9,13,17,20,23 are parents -> slot id; others -1.
constexpr int kParentSlot[L_DIM] = {
     0, -1, -1, -1,  1, -1, -1,  2, -1,  3,
    -1, -1, -1,  4, -1, -1, -1,  5, -1, -1,
     6, -1, -1,  7, -1};
// Column -> slot of its parent (for child columns), -1 otherwise.
constexpr int kChildSlot[L_DIM] = {
    -1,  0,  0,  0, -1,  1,  1, -1,  2, -1,
     3,  3,  3, -1,  4,  4,  4, -1,  5,  5,
    -1,  6,  6, -1,  7};

// Matrix-pipe wave reduction:
// a={acc,0} puts acc[m] at (row m, K=0) for lanes 0-15 and acc[m+16] at
// (row m, K=2) for lanes 16-31. With B = ones(4x16):
//   D[m][n] = acc[m] + acc[m+16]   (same value in every column n)
// D layout: lanes 0-15 hold rows 0-7, lanes 16-31 hold rows 8-15, so
// sum(d[0..7]) gives lanes 0-15: sum(acc[0..7]+acc[16..23]) and
// lanes 16-31: sum(acc[8..15]+acc[24..31]).  Lane0 + lane16 = wave total,
// written to LDS directly -- no shuffle needed.
__device__ __forceinline__ float wave_halfsum_wmma(float acc) {
    v2f a; a.x = acc;  a.y = 0.0f;
    v2f b; b.x = 1.0f; b.y = 1.0f;
    v8f c = {0.f, 0.f, 0.f, 0.f, 0.f, 0.f, 0.f, 0.f};
    v8f d = __builtin_amdgcn_wmma_f32_16x16x4_f32(
        /*neg_a=*/false, a, /*neg_b=*/false, b,
        /*c_mod=*/(short)0, c, /*reuse_a=*/false, /*reuse_b=*/false);
    return ((d[0] + d[1]) + (d[2] + d[3])) + ((d[4] + d[5]) + (d[6] + d[7]));
}

// Per-block: WMMA half-sums -> 2 LDS slots per wave -> thread 0 totals 16.
__device__ __forceinline__ void block_reduce_store(float acc, float* dst,
                                                   float scale) {
    const float hs = wave_halfsum_wmma(acc);   // EXEC all 1s here

    __shared__ float smem[2 * NWAVES];
    const int wid  = threadIdx.x >> 5;
    const int lane = threadIdx.x & 31;
    if (lane == 0)  smem[2 * wid]     = hs;    // rows 0-7 half
    if (lane == 16) smem[2 * wid + 1] = hs;    // rows 8-15 half
    __syncthreads();
    if (threadIdx.x == 0) {
        float tot = 0.0f;
#pragma unroll
        for (int i = 0; i < 2 * NWAVES; ++i) tot += smem[i];
        *dst = tot * scale;
    }
}

__global__ void __launch_bounds__(BLOCK)
focal_main_kernel(const float* __restrict__ logits,
                  const float* __restrict__ targets,
                  float* __restrict__ partial,
                  int rows) {
    const long long row = (long long)blockIdx.x * BLOCK + threadIdx.x;
    float acc = 0.0f;

    if (row < rows) {
        const float* __restrict__ xr = logits  + row * (long long)L_DIM;
        const float* __restrict__ tr = targets + row * (long long)L_DIM;

        float ptg[8];
#pragma unroll
        for (int l = 0; l < L_DIM; ++l) {
            const float x = xr[l];
            const float t = tr[l];
            // bce = max(x,0) - x*t + log1p(exp(-|x|))
            const float u   = __expf(-fabsf(x));
            const float sp  = __logf(1.0f + u);
            const float bce = fmaxf(x, 0.0f) - x * t + sp;
            const float pt  = __expf(-bce);
            const float om  = 1.0f - pt;
            const float focal = 0.75f * om * om * bce;   // ALPHA=.75, GAMMA=2

            if (kParentSlot[l] >= 0) ptg[kParentSlot[l]] = t;  // folds: reg copy

            float mult = 1.0f;                            // HPW=1 -> focal*(1+cond)
            if (kChildSlot[l] >= 0)
                mult += ((x > 0.5f) && (ptg[kChildSlot[l]] < 0.5f)) ? 1.0f : 0.0f;

            const float w = 0.25f + 0.5f * t;  // t*ALPHA + (1-t)*(1-ALPHA)
            acc += focal * mult * w;
        }
    }

    // All threads (including guarded-out rows, acc=0) converge here.
    block_reduce_store(acc, &partial[blockIdx.x], 1.0f);
}

__global__ void __launch_bounds__(BLOCK)
focal_finish_kernel(const float* __restrict__ partial, int nparts,
                    float* __restrict__ out, float inv_total) {
    float acc = 0.0f;
    for (int i = threadIdx.x; i < nparts; i += BLOCK) acc += partial[i];

    block_reduce_store(acc, out, inv_total);
}

extern "C" void kernel_launch(void* const* d_in, const int* in_sizes, int n_in,
                              void* d_out, int out_size, void* d_ws, size_t ws_size,
                              hipStream_t stream) {
    const float* logits  = (const float*)d_in[0];
    const float* targets = (const float*)d_in[1];
    float* out     = (float*)d_out;
    float* partial = (float*)d_ws;

    const int total_elems = in_sizes[0];           // 25,000,000
    const int rows        = total_elems / L_DIM;   // 1,000,000

    const int nblocks = (rows + BLOCK - 1) / BLOCK;   // 3907 blocks, 1 row/thread

    focal_main_kernel<<<nblocks, BLOCK, 0, stream>>>(logits, targets, partial, rows);
    focal_finish_kernel<<<1, BLOCK, 0, stream>>>(partial, nblocks, out,
                                                 1.0f / (float)total_elems);
}